// CTCLoss_33956011442291
// MI455X (gfx1250) — compile-verified
//
#include <hip/hip_runtime.h>
#include <stdint.h>

// ---------------------------------------------------------------------------
// CTC loss for MI455X (gfx1250, wave32).
//   Kernel A: per-(b,t) log-sum-exp stats (streaming, ~268MB read).
//   Kernel B: per-batch forward recursion, logit rows staged into LDS with a
//             depth-6 / 8-buffer GLOBAL_LOAD_ASYNC_TO_LDS_B128 pipeline
//             (s_wait_asynccnt), one barrier per time step, log2-domain math.
// ---------------------------------------------------------------------------

#define LOG2E_F 1.4426950408889634f
#define LN2_F   0.6931471805599453f
#define NEGF    (-1e30f)

// ---- fast base-2 transcendentals (v_exp_f32 / v_log_f32) ------------------
__device__ __forceinline__ float fast_exp2(float x) {
#if __has_builtin(__builtin_amdgcn_exp2f)
    return __builtin_amdgcn_exp2f(x);
#else
    return exp2f(x);
#endif
}
__device__ __forceinline__ float fast_log2(float x) {
#if __has_builtin(__builtin_amdgcn_logf)
    return __builtin_amdgcn_logf(x);
#else
    return __log2f(x);
#endif
}

// ---- CDNA5 async global->LDS path -----------------------------------------
#if __has_builtin(__builtin_amdgcn_global_load_async_to_lds_b128)
#define HAVE_ASYNC_LDS 1
#else
#define HAVE_ASYNC_LDS 0
#endif

#if defined(__HIP_DEVICE_COMPILE__) && !HAVE_ASYNC_LDS
#warning "gfx1250 async-to-LDS builtin not found; falling back to synchronous LDS staging"
#endif

typedef int v4i __attribute__((ext_vector_type(4)));
typedef __attribute__((address_space(1))) v4i* gv4i_ptr;   // global int4*
typedef __attribute__((address_space(3))) v4i* lv4i_ptr;   // LDS int4*

// Copy 16 bytes global -> LDS for this lane.
__device__ __forceinline__ void async_copy_b128(const float* g, float* l) {
#if HAVE_ASYNC_LDS
    __builtin_amdgcn_global_load_async_to_lds_b128(
        (gv4i_ptr)(void*)g, (lv4i_ptr)(void*)l, 0, 0);
#else
    *(float4*)l = *(const float4*)g;   // synchronous fallback
#endif
}

template <int N>
__device__ __forceinline__ void wait_async() {
#if HAVE_ASYNC_LDS
#if __has_builtin(__builtin_amdgcn_s_wait_asynccnt)
    __builtin_amdgcn_s_wait_asynccnt((unsigned short)N);
#else
    asm volatile("s_wait_asynccnt %0" :: "i"(N) : "memory");
#endif
#endif
}

// ---------------------------------------------------------------------------
// Kernel A: stat2[row] = log2( sum_c exp(pred[row,c]) )   (row = b*T + t)
// One wave (32 lanes) per row of C=512 floats; 16 floats per lane (4x float4).
// ---------------------------------------------------------------------------
__global__ void __launch_bounds__(256)
ctc_row_stats(const float* __restrict__ pred, float* __restrict__ stat2, int nrows) {
    constexpr int C = 512;
    const int wave = threadIdx.x >> 5;
    const int lane = threadIdx.x & 31;
    const int row  = blockIdx.x * 8 + wave;
    if (row >= nrows) return;

    const float4* rp = (const float4*)(pred + (size_t)row * C);  // 128 float4 per row
    float4 v[4];
#pragma unroll
    for (int j = 0; j < 4; ++j) v[j] = rp[lane + 32 * j];

    float m = v[0].x;
#pragma unroll
    for (int j = 0; j < 4; ++j)
        m = fmaxf(m, fmaxf(fmaxf(v[j].x, v[j].y), fmaxf(v[j].z, v[j].w)));
#pragma unroll
    for (int off = 16; off >= 1; off >>= 1) m = fmaxf(m, __shfl_xor(m, off));

    float s = 0.f;
#pragma unroll
    for (int j = 0; j < 4; ++j) {
        s += fast_exp2((v[j].x - m) * LOG2E_F);
        s += fast_exp2((v[j].y - m) * LOG2E_F);
        s += fast_exp2((v[j].z - m) * LOG2E_F);
        s += fast_exp2((v[j].w - m) * LOG2E_F);
    }
#pragma unroll
    for (int off = 16; off >= 1; off >>= 1) s += __shfl_xor(s, off);

    if (lane == 0) stat2[row] = m * LOG2E_F + fast_log2(s);
}

// ---------------------------------------------------------------------------
// Kernel B: forward recursion. One workgroup (544 thr = 17 wave32) per batch.
// Depth-6 async prefetch pipeline over 8 LDS row buffers:
//   - buffer (t+6)%8 written at iter t was last read at iter t-2 (2 barriers
//     earlier) -> race-free with a single __syncthreads per step.
//   - only waves 0-3 (tid<128) issue DMA; per-wave in-order completion makes
//     s_wait_asynccnt<=6 guarantee row t is resident.
// ---------------------------------------------------------------------------
__global__ void __launch_bounds__(544)
ctc_alpha(const float* __restrict__ pred, const int* __restrict__ targets,
          const int* __restrict__ in_len, const int* __restrict__ tg_len,
          const float* __restrict__ stat2, float* __restrict__ out,
          int T, int S, int B) {
    constexpr int C    = 512;
    constexpr int MAXL = 513;            // 2*256+1
    constexpr int MAXT = 2048;
    constexpr int NBUF = 8;              // LDS row buffers (16 KB)
    constexpr int PFD  = 6;              // prefetch depth (<= NBUF-2)

    __shared__ float rowbuf[NBUF][C];
    __shared__ float stats_s[MAXT];      // per-t log2-sum-exp (8 KB)
    __shared__ float alpha[2][MAXL + 3]; // ping-pong, 2 leading NEG pads

    const int b   = blockIdx.x;
    const int tid = threadIdx.x;
    const int L   = 2 * S + 1;           // 513 here

    const float* predb = pred  + (size_t)b * T * C;
    const float* statb = stat2 + (size_t)b * T;

    int len = in_len[b];
    if (len < 1) len = 1;
    if (len > T) len = T;
    const int tlen = tg_len[b];

    // Per-thread extended label + skip-transition flag (s-2 -> s).
    int  lab  = 0;
    bool skip = false;
    if (tid < L && (tid & 1)) {
        const int i = tid >> 1;
        lab = targets[(size_t)b * S + i];
        const int prev = (i > 0) ? targets[(size_t)b * S + i - 1] : 0;
        skip = (tid >= 2) && (lab != prev);
    }

    // NEG pads at alpha[*][0..1] (so s-1 / s-2 reads never branch).
    if (tid < 2) { alpha[0][tid] = NEGF; alpha[1][tid] = NEGF; }

    // Prologue DMA (waves 0-3 only): stats row, then rows 0..PFD.
    if (tid < 128) {
#pragma unroll
        for (int j = 0; j < 4; ++j) {
            const int idx = (tid + 128 * j) * 4;
            if (idx < T && idx < MAXT)
                async_copy_b128(statb + idx, &stats_s[idx]);
        }
#pragma unroll
        for (int r = 0; r <= PFD; ++r) {
            if (r < len)
                async_copy_b128(predb + (size_t)r * C + tid * 4,
                                &rowbuf[r % NBUF][tid * 4]);
        }
    }
    // Row 0 + stats resident: in-order => counter<=PFD leaves only rows 1..PFD.
    if (len > PFD) wait_async<PFD>(); else wait_async<0>();
    __syncthreads();

    // t = 0 init: only s=0 (blank) and s=1 reachable.
    if (tid < L) {
        const float p0 = rowbuf[0][lab] * LOG2E_F - stats_s[0];
        alpha[0][tid + 2] = (tid < 2) ? p0 : NEGF;
    }
    int cur = 0;

    for (int t = 1; t < len; ++t) {
        const bool pf = (t + PFD) < len;
        if (pf && tid < C / 4)
            async_copy_b128(predb + (size_t)(t + PFD) * C + tid * 4,
                            &rowbuf[(t + PFD) % NBUF][tid * 4]);
        if (pf) wait_async<PFD>();   // in-order: row t is complete
        else    wait_async<0>();     // tail (last PFD iters): full drain
        __syncthreads();             // row t visible; prev alpha writes visible

        if (tid < L) {
            const float* rb = rowbuf[t % NBUF];
            const float  p  = rb[lab] * LOG2E_F - stats_s[t];
            const float* ac = alpha[cur];
            const float a   = ac[tid + 2];
            const float a1  = ac[tid + 1];
            const float a2  = skip ? ac[tid] : NEGF;
            const float m3  = fmaxf(a, fmaxf(a1, a2));
            const float sum = fast_exp2(a - m3) + fast_exp2(a1 - m3) + fast_exp2(a2 - m3);
            alpha[cur ^ 1][tid + 2] = m3 + fast_log2(sum) + p;
        }
        cur ^= 1;
    }
    __syncthreads();

    if (tid == 0) {
        int end = 2 * tlen;
        if (end > L - 1) end = L - 1;
        const float ae  = alpha[cur][end + 2];
        const float ae1 = (tlen > 0) ? alpha[cur][end + 1] : NEGF;
        const float mm  = fmaxf(ae, ae1);
        const float l2  = mm + fast_log2(fast_exp2(ae - mm) + fast_exp2(ae1 - mm));
        float loss = -(l2 * LN2_F);
        loss = (loss < 1e29f) ? loss : 0.0f;        // zero_infinity
        const float dn = (float)((tlen > 0) ? tlen : 1);
        atomicAdd(out, loss / dn / (float)B);
    }
}

__global__ void ctc_zero_out(float* out) {
    if (threadIdx.x == 0 && blockIdx.x == 0) out[0] = 0.0f;
}

// ---------------------------------------------------------------------------
extern "C" void kernel_launch(void* const* d_in, const int* in_sizes, int n_in,
                              void* d_out, int out_size, void* d_ws, size_t ws_size,
                              hipStream_t stream) {
    (void)n_in; (void)out_size; (void)ws_size;
    const float* pred    = (const float*)d_in[0];   // [B,T,C] fp32
    const int*   targets = (const int*)d_in[1];     // [B,S]
    const int*   in_len  = (const int*)d_in[2];     // [B]
    const int*   tg_len  = (const int*)d_in[3];     // [B]

    const int B = in_sizes[2];
    const int S = in_sizes[1] / B;
    const int C = 512;
    const int T = in_sizes[0] / (B * C);

    float* out   = (float*)d_out;
    float* stat2 = (float*)d_ws;                    // B*T floats (512 KB)
    const int nrows = B * T;

    ctc_zero_out<<<1, 32, 0, stream>>>(out);
    ctc_row_stats<<<(nrows + 7) / 8, 256, 0, stream>>>(pred, stat2, nrows);
    ctc_alpha<<<B, 544, 0, stream>>>(pred, targets, in_len, tg_len, stat2, out, T, S, B);
}